// CustomLayer_opt_30279519437511
// MI455X (gfx1250) — compile-verified
//
#include <hip/hip_runtime.h>
#include <cstdint>

// ---------------- problem constants ----------------
#define NPT   128
#define NN    (NPT*NPT)
#define NNN   (NPT*NPT*NPT)
#define BATCH 2
#define RAD   3
#define VOL   (BATCH*NNN)          // 4,194,304 elements per volume
#define LAMR  0.05f
#define EPSR  1e-8f
#define NITER 8
#define PI_F  3.14159265358979323846f

typedef __attribute__((ext_vector_type(2))) float v2f;
typedef __attribute__((ext_vector_type(8))) float v8f;

// ---------------- DFT twiddle tables ----------------
// cosT[k*128+f] = cos(2*pi*(k*f mod 128)/128); sinT = +sin; sinN = -sin.
__global__ void k_build_tables(float* __restrict__ cosT, float* __restrict__ sinT,
                               float* __restrict__ sinN) {
    int k = blockIdx.x;      // 0..127
    int f = threadIdx.x;     // 0..127
    int m = (k * f) & (NPT - 1);
    float ang = (2.0f * PI_F / (float)NPT) * (float)m;
    float c = cosf(ang), s = sinf(ang);
    cosT[k * NPT + f] = c;
    sinT[k * NPT + f] = s;
    sinN[k * NPT + f] = -s;
}

// ---------------- WMMA DFT pass along one axis ----------------
// Out[line, f] over 128 freqs; element (line,k) lives at
//   addr = (line>>SHIFT)<<(SHIFT+7) | (line & (stride-1)) | k<<SHIFT
// Generic: Re = X*C + Y*S_re ; Im = Y*C + X*S_im
// Forward (e^{-i}): S_re = +sin table, S_im = -sin table
// Inverse (e^{+i}): S_re = -sin table, S_im = +sin table
template<int HIM, int HOIM, int SHIFT>
__global__ __launch_bounds__(128) void dft_pass(
    const float* __restrict__ in_re, const float* __restrict__ in_im,
    float* __restrict__ out_re, float* __restrict__ out_im,
    const float* __restrict__ cosT,
    const float* __restrict__ sinRe, const float* __restrict__ sinIm,
    float scale)
{
    constexpr int stride = 1 << SHIFT;
    const int lane   = threadIdx.x & 31;
    const int wave   = threadIdx.x >> 5;
    const int hi     = lane >> 4;          // lane group (0: lanes 0-15, 1: 16-31)
    const int fl     = lane & 15;
    const int lineBase = (blockIdx.x * 4 + wave) * 16;   // uniform per wave

    // A-matrix source row for this lane (M = fl in both lane halves)
    const int lineA = lineBase + fl;
    const int abase = ((lineA >> SHIFT) << (SHIFT + 7)) + (lineA & (stride - 1));

    // C/D row bases: VGPR i holds rows M=i (lanes 0-15) and M=i+8 (lanes 16-31)
    int rb[8];
#pragma unroll
    for (int i = 0; i < 8; ++i) {
        int l2 = lineBase + i + hi * 8;
        rb[i] = ((l2 >> SHIFT) << (SHIFT + 7)) + (l2 & (stride - 1));
    }

    v8f zero = {};
    v8f accRe[8];
    v8f accIm[HOIM ? 8 : 1];
#pragma unroll
    for (int t = 0; t < 8; ++t) {
        accRe[t] = zero;
        if constexpr (HOIM) accIm[t] = zero;
    }

    const int kA = 2 * hi;   // A layout: VGPR0={K0,K2}, VGPR1={K1,K3} across lane halves
    for (int k0 = 0; k0 < NPT; k0 += 4) {
        const int ki = k0 + kA;
        v2f ax, ay;
        ax.x = in_re[abase + (ki << SHIFT)];
        ax.y = in_re[abase + ((ki + 1) << SHIFT)];
        if constexpr (HIM) {
            ay.x = in_im[abase + (ki << SHIFT)];
            ay.y = in_im[abase + ((ki + 1) << SHIFT)];
        }
        // Unconditional speculative prefetch of the next k-chunk
        // (gfx1250 global_prefetch_b8; invalid translations are dropped).
        __builtin_prefetch(in_re + abase + ((ki + 4) << SHIFT), 0, 1);
        if constexpr (HIM)
            __builtin_prefetch(in_im + abase + ((ki + 4) << SHIFT), 0, 1);

        const float* cp  = cosT  + ki * NPT + fl;
        const float* spr = sinRe + ki * NPT + fl;
        const float* spi = sinIm + ki * NPT + fl;
#pragma unroll
        for (int ft = 0; ft < 8; ++ft) {
            const int fo = ft * 16;
            v2f bc;   // B layout: VGPR0={K0,K2}, VGPR1={K1,K3}
            bc.x = cp[fo]; bc.y = cp[fo + NPT];
            accRe[ft] = __builtin_amdgcn_wmma_f32_16x16x4_f32(
                false, ax, false, bc, (short)0, accRe[ft], false, false);
            if constexpr (HIM) {
                v2f bsr;
                bsr.x = spr[fo]; bsr.y = spr[fo + NPT];
                accRe[ft] = __builtin_amdgcn_wmma_f32_16x16x4_f32(
                    false, ay, false, bsr, (short)0, accRe[ft], false, false);
            }
            if constexpr (HOIM) {
                if constexpr (HIM) {
                    accIm[ft] = __builtin_amdgcn_wmma_f32_16x16x4_f32(
                        false, ay, false, bc, (short)0, accIm[ft], false, false);
                }
                v2f bsi;
                bsi.x = spi[fo]; bsi.y = spi[fo + NPT];
                accIm[ft] = __builtin_amdgcn_wmma_f32_16x16x4_f32(
                    false, ax, false, bsi, (short)0, accIm[ft], false, false);
            }
        }
    }

#pragma unroll
    for (int ft = 0; ft < 8; ++ft) {
        const int fofs = (fl + ft * 16) << SHIFT;
#pragma unroll
        for (int i = 0; i < 8; ++i) {
            out_re[rb[i] + fofs] = accRe[ft][i] * scale;
            if constexpr (HOIM) out_im[rb[i] + fofs] = accIm[ft][i] * scale;
        }
    }
}

// ---------------- pointwise / CG kernels ----------------
__global__ void k_kmul(const float* __restrict__ fre, const float* __restrict__ fim,
                       const float* __restrict__ smv_r,
                       float* __restrict__ ore, float* __restrict__ oim, int accFlag) {
    int i = blockIdx.x * blockDim.x + threadIdx.x;
    if (i >= VOL) return;
    float s  = smv_r[i & (NNN - 1)];
    float vr = s * fre[i], vi = s * fim[i];
    if (accFlag) { ore[i] += vr; oim[i] += vi; }
    else         { ore[i]  = vr; oim[i]  = vi; }
}

__global__ void k_maskw(float* __restrict__ wm, const float* __restrict__ masks,
                        const float* __restrict__ weight, int r) {
    int i = blockIdx.x * blockDim.x + threadIdx.x;
    if (i >= VOL) return;
    wm[i] = wm[i] * masks[i * RAD + r] * weight[i];
}

__global__ void k_apply_lam(const float* __restrict__ back, const float* __restrict__ p,
                            float* __restrict__ Ap) {
    int i = blockIdx.x * blockDim.x + threadIdx.x;
    if (i >= VOL) return;
    Ap[i] = back[i] + LAMR * p[i];
}

__global__ void k_resid(const float* __restrict__ rhs, const float* __restrict__ Ap,
                        float* __restrict__ rv, float* __restrict__ pv) {
    int i = blockIdx.x * blockDim.x + threadIdx.x;
    if (i >= VOL) return;
    float t = rhs[i] - Ap[i];
    rv[i] = t; pv[i] = t;
}

__global__ void k_update_xr(float* __restrict__ xk, float* __restrict__ rv,
                            const float* __restrict__ pv, const float* __restrict__ Ap,
                            const float* __restrict__ scal) {
    int i = blockIdx.x * blockDim.x + threadIdx.x;
    if (i >= VOL) return;
    float alpha = scal[2];
    xk[i] += alpha * pv[i];
    rv[i] -= alpha * Ap[i];
}

__global__ void k_update_p(float* __restrict__ pv, const float* __restrict__ rv,
                           const float* __restrict__ scal) {
    int i = blockIdx.x * blockDim.x + threadIdx.x;
    if (i >= VOL) return;
    float beta = scal[3];
    pv[i] = rv[i] + beta * pv[i];
}

// deterministic two-stage dot product (fixed reduction order, no atomics)
__global__ __launch_bounds__(256) void k_dot_partial(const float* __restrict__ a,
                                                     const float* __restrict__ b,
                                                     float* __restrict__ partials) {
    __shared__ float sm[256];
    float s = 0.f;
    for (int i = blockIdx.x * 256 + threadIdx.x; i < VOL; i += 256 * 256)
        s += a[i] * b[i];
    sm[threadIdx.x] = s;
    __syncthreads();
    for (int off = 128; off > 0; off >>= 1) {
        if (threadIdx.x < off) sm[threadIdx.x] += sm[threadIdx.x + off];
        __syncthreads();
    }
    if (threadIdx.x == 0) partials[blockIdx.x] = sm[0];
}

// mode 0: rs = sum        (scal[0])
// mode 1: pAp = sum; alpha = rs/(pAp+eps)          (scal[1], scal[2])
// mode 2: rs_new = sum; beta = rs_new/(rs+eps); rs = rs_new  (scal[4], scal[3], scal[0])
__global__ void k_fin(const float* __restrict__ partials, float* __restrict__ scal, int mode) {
    if (threadIdx.x != 0 || blockIdx.x != 0) return;
    float s = 0.f;
    for (int i = 0; i < 256; ++i) s += partials[i];
    if (mode == 0) {
        scal[0] = s;
    } else if (mode == 1) {
        scal[1] = s;
        scal[2] = scal[0] / (s + EPSR);
    } else {
        scal[4] = s;
        scal[3] = s / (scal[0] + EPSR);
        scal[0] = s;
    }
}

// ---------------- host-side orchestration ----------------
namespace {

constexpr int EW_GRID = VOL / 256;            // elementwise grid
constexpr int DFT_GRID = (VOL / NPT) / 64;    // 32768 lines, 64 lines/block = 512

struct Ws {
    float *cosT, *sinT, *sinN, *scal, *partials;
    float *c1re, *c1im, *c2re, *c2im, *c3re, *c3im;
    float *Fvre, *Fvim, *accRe, *accIm;
    float *wm, *Ap, *xk, *rv, *pv;
};

inline Ws layout(void* d_ws) {
    float* w = (float*)d_ws;
    Ws s;
    s.cosT = w;             s.sinT = w + 16384;  s.sinN = w + 32768;
    s.scal = w + 49152;     s.partials = w + 49168;
    float* vb = w + 49664;  // 256B-aligned volume base
    auto V = [&](int i) { return vb + (size_t)i * VOL; };
    s.c1re = V(0);  s.c1im = V(1);  s.c2re = V(2);  s.c2im = V(3);
    s.c3re = V(4);  s.c3im = V(5);  s.Fvre = V(6);  s.Fvim = V(7);
    s.accRe = V(8); s.accIm = V(9); s.wm = V(10);   s.Ap = V(11);
    s.xk = V(12);   s.rv = V(13);   s.pv = V(14);
    return s;
}

// forward: S_re = +sin, S_im = -sin
inline void fwd3_from_real(const float* rin, float* t1re, float* t1im,
                           float* t2re, float* t2im, float* dre, float* dim,
                           const Ws& w, hipStream_t s) {
    dft_pass<0,1,0> <<<DFT_GRID,128,0,s>>>(rin, nullptr, t1re, t1im,
                                           w.cosT, w.sinT, w.sinN, 1.0f);
    dft_pass<1,1,7> <<<DFT_GRID,128,0,s>>>(t1re, t1im, t2re, t2im,
                                           w.cosT, w.sinT, w.sinN, 1.0f);
    dft_pass<1,1,14><<<DFT_GRID,128,0,s>>>(t2re, t2im, dre, dim,
                                           w.cosT, w.sinT, w.sinN, 1.0f);
}

// inverse: S_re = -sin, S_im = +sin; 1/N per pass; last pass real-only output
inline void inv3_to_real(const float* sre, const float* sim,
                         float* t1re, float* t1im, float* t2re, float* t2im,
                         float* rout, const Ws& w, hipStream_t s) {
    const float sc = 1.0f / (float)NPT;
    dft_pass<1,1,0> <<<DFT_GRID,128,0,s>>>(sre, sim, t1re, t1im,
                                           w.cosT, w.sinN, w.sinT, sc);
    dft_pass<1,1,7> <<<DFT_GRID,128,0,s>>>(t1re, t1im, t2re, t2im,
                                           w.cosT, w.sinN, w.sinT, sc);
    dft_pass<1,0,14><<<DFT_GRID,128,0,s>>>(t2re, t2im, rout, nullptr,
                                           w.cosT, w.sinN, w.sinT, sc);
}

// Ap = sum_r SMV_r^T (m_r * w * SMV_r(src)) + LAM*src
inline void applyA(const float* src, const float* masks, const float* weight,
                   const float* smv, const Ws& w, hipStream_t s) {
    fwd3_from_real(src, w.c1re, w.c1im, w.c2re, w.c2im, w.Fvre, w.Fvim, w, s);
    for (int r = 0; r < RAD; ++r) {
        const float* smv_r = smv + (size_t)r * NNN;
        k_kmul<<<EW_GRID,256,0,s>>>(w.Fvre, w.Fvim, smv_r, w.c1re, w.c1im, 0);
        inv3_to_real(w.c1re, w.c1im, w.c2re, w.c2im, w.c3re, w.c3im, w.wm, w, s);
        k_maskw<<<EW_GRID,256,0,s>>>(w.wm, masks, weight, r);
        fwd3_from_real(w.wm, w.c2re, w.c2im, w.c3re, w.c3im, w.c1re, w.c1im, w, s);
        k_kmul<<<EW_GRID,256,0,s>>>(w.c1re, w.c1im, smv_r, w.accRe, w.accIm, r > 0 ? 1 : 0);
    }
    inv3_to_real(w.accRe, w.accIm, w.c1re, w.c1im, w.c2re, w.c2im, w.wm, w, s);
    k_apply_lam<<<EW_GRID,256,0,s>>>(w.wm, src, w.Ap);
}

} // namespace

extern "C" void kernel_launch(void* const* d_in, const int* in_sizes, int n_in,
                              void* d_out, int out_size, void* d_ws, size_t ws_size,
                              hipStream_t stream) {
    const float* rhs    = (const float*)d_in[0];  // x        [B,D,H,W,1]
    const float* masks  = (const float*)d_in[1];  // masks    [B,D,H,W,R]
    const float* weight = (const float*)d_in[2];  // weight   [B,D,H,W,1]
    const float* smv    = (const float*)d_in[3];  // smv      [R,D,H,W]
    const float* init_x = (const float*)d_in[4];  // init_x   [B,D,H,W,1]
    float* out = (float*)d_out;

    Ws w = layout(d_ws);

    k_build_tables<<<NPT,NPT,0,stream>>>(w.cosT, w.sinT, w.sinN);

    // x0 = init_x; r = p = rhs - A(x0); rs = <r,r>
    (void)hipMemcpyAsync(w.xk, init_x, (size_t)VOL * sizeof(float),
                         hipMemcpyDeviceToDevice, stream);
    applyA(w.xk, masks, weight, smv, w, stream);
    k_resid<<<EW_GRID,256,0,stream>>>(rhs, w.Ap, w.rv, w.pv);
    k_dot_partial<<<256,256,0,stream>>>(w.rv, w.rv, w.partials);
    k_fin<<<1,1,0,stream>>>(w.partials, w.scal, 0);

    for (int it = 0; it < NITER; ++it) {
        applyA(w.pv, masks, weight, smv, w, stream);
        k_dot_partial<<<256,256,0,stream>>>(w.pv, w.Ap, w.partials);
        k_fin<<<1,1,0,stream>>>(w.partials, w.scal, 1);           // alpha
        k_update_xr<<<EW_GRID,256,0,stream>>>(w.xk, w.rv, w.pv, w.Ap, w.scal);
        k_dot_partial<<<256,256,0,stream>>>(w.rv, w.rv, w.partials);
        k_fin<<<1,1,0,stream>>>(w.partials, w.scal, 2);           // beta, rs
        k_update_p<<<EW_GRID,256,0,stream>>>(w.pv, w.rv, w.scal);
    }

    (void)hipMemcpyAsync(out, w.xk, (size_t)VOL * sizeof(float),
                         hipMemcpyDeviceToDevice, stream);
}